// GCContext_74242804678700
// MI455X (gfx1250) — compile-verified
//
#include <hip/hip_runtime.h>

// CDNA5 / gfx1250. wave32. Memory-bound problem (512MB traffic @ 23.3TB/s ~ 22us),
// fp32 kept end-to-end; channel contraction fused via V_WMMA_F32_16X16X4_F32.
// attn staging in kernel 3 uses GLOBAL_LOAD_ASYNC_TO_LDS (ASYNCcnt).

typedef __attribute__((ext_vector_type(2))) float v2f;
typedef __attribute__((ext_vector_type(8))) float v8f;

// Types for the async global->LDS builtin: b128 = <4 x i32>, AS1 src, AS3 dst.
typedef int v4i_t __attribute__((vector_size(16)));
typedef v4i_t __attribute__((address_space(1))) as1_v4i;
typedef v4i_t __attribute__((address_space(3))) as3_v4i;

#define C_DIM 1024
#define S_DIM 4096   // 64*64
#define B_DIM 16

// ---------------------------------------------------------------------------
// Kernel 1: logits[b,s] = (sum_c x[b,c,s]*w[c] + bias) * (1/sqrt(C))
// One wave handles 16 spatial positions via WMMA f32 16x16x4:
//   A[m][k] = x[b, k0+k, s0+m]   (M = spatial across lanes -> contiguous loads)
//   B[k][n] = w[k0+k]            (replicated over N, from LDS)
//   D[m][n] = partial logit for s0+m, identical over n; accumulate over k0.
// ---------------------------------------------------------------------------
__global__ __launch_bounds__(128) void logits_wmma_kernel(
    const float* __restrict__ x, const float* __restrict__ w,
    const float* __restrict__ bptr, float* __restrict__ logits) {
  __shared__ float wlds[C_DIM];
  const int tid = threadIdx.x;
  #pragma unroll
  for (int i = tid; i < C_DIM; i += 128) wlds[i] = w[i];
  __syncthreads();

  const int b    = blockIdx.x >> 6;   // 64 blocks per batch
  const int tile = blockIdx.x & 63;   // 64 spatial per block (4 waves x 16)
  const int wave = tid >> 5;
  const int lane = tid & 31;
  const int half = lane >> 4;         // 0: K={0,1}, 1: K={2,3}
  const int m    = lane & 15;         // M (spatial) index within tile
  const int sel  = half << 1;
  const int s0   = (tile << 6) + (wave << 4);

  // x[b, c, s] : c-stride = 4096 floats, batch stride = 1024*4096 floats
  const float* __restrict__ px = x + ((long)b << 22) + (long)(s0 + m);

  v8f acc = {};   // 16x16 f32 accumulator (all N columns identical)
  #pragma unroll 4
  for (int k0 = 0; k0 < C_DIM; k0 += 4) {
    const int ks = k0 + sel;
    const float a0 = px[(long)ks * S_DIM];
    const float a1 = px[(long)(ks + 1) * S_DIM];
    v2f A = {a0, a1};                          // A[m][sel], A[m][sel+1]
    v2f B = {wlds[ks], wlds[ks + 1]};          // B[sel][*], B[sel+1][*]
    acc = __builtin_amdgcn_wmma_f32_16x16x4_f32(
        /*neg_a=*/false, A, /*neg_b=*/false, B,
        /*c_mod=*/(short)0, acc, /*reuse_a=*/false, /*reuse_b=*/false);
  }

  const float bias  = bptr[0];
  const float scale = 0.03125f;  // 1/sqrt(1024)
  // D layout: acc[j] on lane (half*16 + n) is D[half*8 + j][n]; pick lane n==j.
  float* __restrict__ lp = logits + b * S_DIM + s0 + (half << 3);
  #pragma unroll
  for (int j = 0; j < 8; ++j) {
    if (m == j) lp[j] = (acc[j] + bias) * scale;
  }
}

// ---------------------------------------------------------------------------
// Kernel 2: per-batch softmax over 4096 logits. One 256-thread block per batch.
// ---------------------------------------------------------------------------
__global__ __launch_bounds__(256) void softmax_kernel(
    const float* __restrict__ logits, float* __restrict__ attn) {
  __shared__ float red[256];
  const int b   = blockIdx.x;
  const int tid = threadIdx.x;
  const float* __restrict__ lp = logits + b * S_DIM;

  float v[16];
  float mx = -3.402823466e+38f;
  #pragma unroll
  for (int i = 0; i < 16; ++i) {
    v[i] = lp[tid + (i << 8)];
    mx = fmaxf(mx, v[i]);
  }
  red[tid] = mx; __syncthreads();
  for (int off = 128; off > 0; off >>= 1) {
    if (tid < off) red[tid] = fmaxf(red[tid], red[tid + off]);
    __syncthreads();
  }
  mx = red[0]; __syncthreads();

  float sum = 0.f;
  #pragma unroll
  for (int i = 0; i < 16; ++i) { v[i] = __expf(v[i] - mx); sum += v[i]; }
  red[tid] = sum; __syncthreads();
  for (int off = 128; off > 0; off >>= 1) {
    if (tid < off) red[tid] += red[tid + off];
    __syncthreads();
  }
  const float inv = 1.0f / red[0];
  float* __restrict__ ap = attn + b * S_DIM;
  #pragma unroll
  for (int i = 0; i < 16; ++i) ap[tid + (i << 8)] = v[i] * inv;
}

// ---------------------------------------------------------------------------
// Kernel 3: focus[b,c] = sum_s x[b,c,s]*attn[b,s].
// Spatial is contiguous + contraction dim -> float4 loads, lanes across
// spatial, wave shfl_xor reduction. attn (16KB/batch) staged in LDS, shared
// by 8 waves; staging uses the gfx1250 async global->LDS engine if present.
// ---------------------------------------------------------------------------
__global__ __launch_bounds__(256) void focus_kernel(
    const float* __restrict__ x, const float* __restrict__ attn,
    float* __restrict__ out) {
  __shared__ float alds[S_DIM];
  const int b    = blockIdx.x >> 7;   // 128 blocks per batch
  const int cblk = blockIdx.x & 127;  // 8 channels per block (1 per wave)
  const int tid  = threadIdx.x;

#if __has_builtin(__builtin_amdgcn_global_load_async_to_lds_b128)
  {
    // 256 lanes x 4 x 16B = 16KB, direct global->LDS, no VGPR staging.
    const float* g = attn + b * S_DIM + tid * 4;
    float* l = alds + tid * 4;
    #pragma unroll
    for (int i = 0; i < 4; ++i) {
      __builtin_amdgcn_global_load_async_to_lds_b128(
          (as1_v4i*)(g + i * 1024),
          (as3_v4i*)(l + i * 1024),
          /*offset=*/0, /*cpol=*/0);
    }
#if __has_builtin(__builtin_amdgcn_s_wait_asynccnt)
    __builtin_amdgcn_s_wait_asynccnt(0);
#else
    asm volatile("s_wait_asynccnt 0x0" ::: "memory");
#endif
  }
#else
  {
    const float4* __restrict__ asrc = (const float4*)(attn + b * S_DIM);
    float4* adst = (float4*)alds;
    #pragma unroll
    for (int i = tid; i < S_DIM / 4; i += 256) adst[i] = asrc[i];
  }
#endif
  __syncthreads();

  const int wave = tid >> 5, lane = tid & 31;
  const int c = (cblk << 3) + wave;
  const float4* __restrict__ xp =
      (const float4*)(x + ((long)b << 22) + (long)c * S_DIM);
  const float4* __restrict__ ap = (const float4*)alds;

  float acc = 0.f;
  #pragma unroll 4
  for (int i = 0; i < 32; ++i) {
    const int s = (i << 5) + lane;     // 128-float stride per step, coalesced
    const float4 xv = xp[s];
    const float4 av = ap[s];
    acc = fmaf(xv.x, av.x, acc);
    acc = fmaf(xv.y, av.y, acc);
    acc = fmaf(xv.z, av.z, acc);
    acc = fmaf(xv.w, av.w, acc);
  }
  #pragma unroll
  for (int off = 16; off > 0; off >>= 1) acc += __shfl_xor(acc, off, 32);
  if (lane == 0) out[b * C_DIM + c] = acc;
}

// ---------------------------------------------------------------------------
extern "C" void kernel_launch(void* const* d_in, const int* in_sizes, int n_in,
                              void* d_out, int out_size, void* d_ws, size_t ws_size,
                              hipStream_t stream) {
  (void)in_sizes; (void)n_in; (void)out_size; (void)ws_size;
  const float* x    = (const float*)d_in[0];  // [16,1024,64,64] fp32
  const float* w    = (const float*)d_in[1];  // [1024]
  const float* bias = (const float*)d_in[2];  // [1]
  float* out = (float*)d_out;                 // [16,1024]

  float* logits = (float*)d_ws;               // 16*4096 floats (256 KB)
  float* attn   = logits + B_DIM * S_DIM;     // 16*4096 floats (256 KB)

  logits_wmma_kernel<<<B_DIM * 64, 128, 0, stream>>>(x, w, bias, logits);
  softmax_kernel<<<B_DIM, 256, 0, stream>>>(logits, attn);
  focus_kernel<<<B_DIM * 128, 256, 0, stream>>>(x, attn, out);
}